// DRModel_1726576853556
// MI455X (gfx1250) — compile-verified
//
#include <hip/hip_runtime.h>
#include <cstdint>

typedef __attribute__((ext_vector_type(2))) float v2f;
typedef __attribute__((ext_vector_type(8))) float v8f;

constexpr int B = 256, S = 50, NI = 40, D = 256;
constexpr int G = 3 * D;         // 768 gate columns
constexpr int M = B * S;         // 12800 GEMM rows
constexpr int LDS_STRIDE = D + 4; // 260: bank = (4*row + col) % 64 -> conflict-free

__device__ __forceinline__ float sigmoidf(float v) {
    return 1.0f / (1.0f + __expf(-v));
}

// ---------------- Kernel 1: embedding gather + masked mean pool ----------------
// grid = B*S blocks, 256 threads (one per feature d). Coalesced row reads; the
// 102 MB embed table is L2-resident on MI455X (192 MB L2).
__global__ void pool_kernel(const int* __restrict__ x,
                            const int* __restrict__ basket,
                            const float* __restrict__ embed,
                            float* __restrict__ pooled) {
    __shared__ int sidx[NI];
    const int bs = blockIdx.x;                 // flattened (b, s)
    const int n = basket[bs];
    if (threadIdx.x < NI) sidx[threadIdx.x] = x[bs * NI + threadIdx.x];
    __syncthreads();
    const int d = threadIdx.x;
    float acc = 0.f;
    for (int i = 0; i < n; ++i)
        acc += embed[(size_t)sidx[i] * D + d];
    pooled[(size_t)bs * D + d] = acc * (1.0f / (float)n);
}

// ---------------- Kernel 2: gates_x = pooled @ w_ih^T + b_ih ----------------
// f32 WMMA 16x16x4. Block = one 16-row M-tile; A panel staged in LDS once
// (coalesced), 8 waves x 6 N-tiles cover all 48 column tiles (G = 768).
// Per k-step per wave: 1 LDS v2f read (A, reused 6x) + 6 global v2f (B) + 6 WMMA.
__global__ void gates_gemm_kernel(const float* __restrict__ A,    // (M, D)
                                  const float* __restrict__ Wih,  // (G, D)
                                  const float* __restrict__ bih,  // (G)
                                  float* __restrict__ gates) {    // (M, G)
    __shared__ float As[16][LDS_STRIDE];
    const int m0 = blockIdx.x * 16;

    for (int i = threadIdx.x; i < 16 * D; i += 256)
        As[i >> 8][i & (D - 1)] = A[(size_t)(m0 + (i >> 8)) * D + (i & (D - 1))];
    __syncthreads();

    const int lane = threadIdx.x & 31;
    const int wave = threadIdx.x >> 5;          // 0..7
    const int lo = lane & 15;
    const int hi = lane >> 4;                   // selects K pair {0,1} vs {2,3}
    const int tn0 = wave * 6;                   // first of 6 N-tiles

    v8f c[6] = {v8f{}, v8f{}, v8f{}, v8f{}, v8f{}, v8f{}};
    for (int k = 0; k < D; k += 4) {
        const int kk = k + 2 * hi;
        const v2f a = *(const v2f*)(&As[lo][kk]);
#pragma unroll
        for (int t = 0; t < 6; ++t) {
            const int col = (tn0 + t) * 16 + lo;
            const v2f b = *(const v2f*)(Wih + (size_t)col * D + kk);
            c[t] = __builtin_amdgcn_wmma_f32_16x16x4_f32(false, a, false, b,
                                                         (short)0, c[t], false, false);
        }
    }
#pragma unroll
    for (int t = 0; t < 6; ++t) {
        const int col = (tn0 + t) * 16 + lo;
        const float bias = bih[col];
#pragma unroll
        for (int j = 0; j < 8; ++j) {
            const int r = m0 + j + 8 * hi;      // C layout: M = j + 8*hi, N = lo
            gates[(size_t)r * G + col] = c[t][j] + bias;
        }
    }
}

// ---------------- Kernel 3: one GRU step ----------------
// grid = B/16 blocks x 256 threads (8 waves). Block owns 16 rows of h, staged
// into LDS once (coalesced). Wave w owns d-columns [32w, 32w+32): six 16x16
// WMMA tiles (2 col-subtiles x {r,z,n}) so gate combination stays in registers.
// All h reads come from the LDS snapshot, so global h can be written directly.
__global__ void gru_step_kernel(float* __restrict__ h,            // (B, D)
                                const float* __restrict__ gates,  // (B*S, G)
                                const float* __restrict__ Whh,    // (G, D)
                                const float* __restrict__ bhh,    // (G)
                                const int* __restrict__ lengths,  // (B)
                                float* __restrict__ out_dyn,      // (B, S, D)
                                int s) {
    __shared__ float Hs[16][LDS_STRIDE];
    const int b0 = blockIdx.x * 16;

    for (int i = threadIdx.x; i < 16 * D; i += 256)
        Hs[i >> 8][i & (D - 1)] = h[(size_t)(b0 + (i >> 8)) * D + (i & (D - 1))];
    __syncthreads();

    const int lane = threadIdx.x & 31;
    const int wave = threadIdx.x >> 5;          // 0..7
    const int lo = lane & 15;
    const int hi = lane >> 4;
    const int d0 = wave * 32;

    v8f cr[2] = {v8f{}, v8f{}};
    v8f cz[2] = {v8f{}, v8f{}};
    v8f cn[2] = {v8f{}, v8f{}};

    for (int k = 0; k < D; k += 4) {
        const int kk = k + 2 * hi;
        const v2f a = *(const v2f*)(&Hs[lo][kk]);
#pragma unroll
        for (int t = 0; t < 2; ++t) {
            const int n = d0 + t * 16 + lo;
            const v2f br = *(const v2f*)(Whh + (size_t)n * D + kk);
            const v2f bz = *(const v2f*)(Whh + (size_t)(D + n) * D + kk);
            const v2f bn = *(const v2f*)(Whh + (size_t)(2 * D + n) * D + kk);
            cr[t] = __builtin_amdgcn_wmma_f32_16x16x4_f32(false, a, false, br,
                                                          (short)0, cr[t], false, false);
            cz[t] = __builtin_amdgcn_wmma_f32_16x16x4_f32(false, a, false, bz,
                                                          (short)0, cz[t], false, false);
            cn[t] = __builtin_amdgcn_wmma_f32_16x16x4_f32(false, a, false, bn,
                                                          (short)0, cn[t], false, false);
        }
    }

#pragma unroll
    for (int t = 0; t < 2; ++t) {
        const int d = d0 + t * 16 + lo;
        const float br_b = bhh[d];
        const float bz_b = bhh[D + d];
        const float bn_b = bhh[2 * D + d];
#pragma unroll
        for (int j = 0; j < 8; ++j) {
            const int row = j + 8 * hi;         // C layout row within tile
            const int b = b0 + row;
            const size_t gbase = ((size_t)b * S + s) * G;
            const float gxr = gates[gbase + d];
            const float gxz = gates[gbase + D + d];
            const float gxn = gates[gbase + 2 * D + d];
            const float hprev = Hs[row][d];
            const float r = sigmoidf(gxr + cr[t][j] + br_b);
            const float z = sigmoidf(gxz + cz[t][j] + bz_b);
            const float nn = tanhf(gxn + r * (cn[t][j] + bn_b));
            const float hnew = (1.0f - z) * nn + z * hprev;
            const bool valid = s < lengths[b];
            h[(size_t)b * D + d] = valid ? hnew : hprev;
            out_dyn[((size_t)b * S + s) * D + d] = valid ? hnew : 0.0f;
        }
    }
}

// ---------------- Kernel 4: emit final hidden state h_u ----------------
__global__ void copy_kernel(const float* __restrict__ src, float* __restrict__ dst, int n) {
    const int i = blockIdx.x * blockDim.x + threadIdx.x;
    if (i < n) dst[i] = src[i];
}

extern "C" void kernel_launch(void* const* d_in, const int* in_sizes, int n_in,
                              void* d_out, int out_size, void* d_ws, size_t ws_size,
                              hipStream_t stream) {
    (void)in_sizes; (void)n_in; (void)out_size; (void)ws_size;

    const int*   x       = (const int*)  d_in[0];
    const int*   basket  = (const int*)  d_in[1];
    const int*   lengths = (const int*)  d_in[2];
    const float* hidden  = (const float*)d_in[3];
    const float* embed   = (const float*)d_in[4];
    const float* w_ih    = (const float*)d_in[5];
    const float* w_hh    = (const float*)d_in[6];
    const float* b_ih    = (const float*)d_in[7];
    const float* b_hh    = (const float*)d_in[8];

    float* pooled  = (float*)d_ws;                   // M * D
    float* gates   = pooled + (size_t)M * D;         // M * G
    float* h       = gates + (size_t)M * G;          // B * D
    float* out_dyn = (float*)d_out;                  // B * S * D
    float* out_h   = out_dyn + (size_t)B * S * D;    // B * D

    hipMemcpyAsync(h, hidden, (size_t)B * D * sizeof(float),
                   hipMemcpyDeviceToDevice, stream);

    pool_kernel<<<M, 256, 0, stream>>>(x, basket, embed, pooled);

    gates_gemm_kernel<<<M / 16, 256, 0, stream>>>(pooled, w_ih, b_ih, gates);

    for (int s = 0; s < S; ++s)
        gru_step_kernel<<<B / 16, 256, 0, stream>>>(h, gates, w_hh, b_hh,
                                                    lengths, out_dyn, s);

    copy_kernel<<<(B * D + 255) / 256, 256, 0, stream>>>(h, out_h, B * D);
}